// GAT_FLEPE_35270271435479
// MI455X (gfx1250) — compile-verified
//
#include <hip/hip_runtime.h>
#include <math.h>

typedef __attribute__((ext_vector_type(2))) float v2f;
typedef __attribute__((ext_vector_type(8))) float v8f;

#define N_NODES 50000
#define N_EDGES 1600000
#define IN_CH   128
#define HID_CH  128
#define OUT_CH  64
#define EDGE_DIM 16
#define NEG_SLOPE 0.2f

// ---------------------------------------------------------------- utilities
__global__ void fill_kernel(float* __restrict__ p, float v, int n) {
    int i = blockIdx.x * blockDim.x + threadIdx.x;
    if (i < n) p[i] = v;
}

// wae[j] = sum_c We[j*C + c] * ae[c]   (collapses  (flepe@We)@ae  ->  flepe@wae)
__global__ void small_matvec_kernel(const float* __restrict__ We,
                                    const float* __restrict__ ae,
                                    float* __restrict__ wae, int C) {
    int j = threadIdx.x;
    if (j >= EDGE_DIM) return;
    float acc = 0.f;
    for (int c = 0; c < C; ++c) acc += We[j * C + c] * ae[c];
    wae[j] = acc;
}

// ------------------------------------------------------- fp32 WMMA GEMM
// H[N,C] = X[N,K] @ W[K,C], exact fp32 via v_wmma_f32_16x16x4_f32.
// One wave computes one 16x16 output tile; blockDim = (32,4).
__global__ void gemm_wmma_f32(const float* __restrict__ X,
                              const float* __restrict__ W,
                              float* __restrict__ H, int K, int C) {
    const int lane   = threadIdx.x;                       // 0..31
    const int tile_m = blockIdx.x;                        // N/16 tiles
    const int tile_n = blockIdx.y * blockDim.y + threadIdx.y;
    const int row    = tile_m * 16 + (lane & 15);
    const int col    = tile_n * 16 + (lane & 15);
    const int koff   = (lane >> 4) << 1;                  // lanes 16-31 hold K+2,K+3

    v8f acc = {};
    const float* xrow = X + (size_t)row * K;
    for (int k = 0; k < K; k += 4) {
        v2f a = *(const v2f*)(xrow + k + koff);           // A: row-major, 8B aligned
        v2f b;
        b.x = W[(size_t)(k + koff)     * C + col];        // B: K-striped across lanes
        b.y = W[(size_t)(k + koff + 1) * C + col];
        acc = __builtin_amdgcn_wmma_f32_16x16x4_f32(
            false, a, false, b, (short)0, acc, false, false);
    }
    // C/D layout: VGPR r -> M = r + 8*(lane>=16), N = lane&15
    const int mbase = tile_m * 16 + ((lane >> 4) << 3);
#pragma unroll
    for (int r = 0; r < 8; ++r)
        H[(size_t)(mbase + r) * C + col] = acc[r];
}

// --------------------------------------------- per-node attention scalars
// alpha_src[i] = h[i,:]·a_s ; alpha_dst[i] = h[i,:]·a_d   (one wave32 / node)
__global__ void node_alpha_kernel(const float* __restrict__ H,
                                  const float* __restrict__ aw_s,
                                  const float* __restrict__ aw_d,
                                  float* __restrict__ as_out,
                                  float* __restrict__ ad_out, int C) {
    int node = blockIdx.x * (blockDim.x >> 5) + (threadIdx.x >> 5);
    int lane = threadIdx.x & 31;
    if (node >= N_NODES) return;
    const float* h = H + (size_t)node * C;
    float s = 0.f, d = 0.f;
    for (int c = lane; c < C; c += 32) {
        float hv = h[c];
        s += hv * aw_s[c];
        d += hv * aw_d[c];
    }
#pragma unroll
    for (int off = 16; off; off >>= 1) {
        s += __shfl_down(s, off, 32);
        d += __shfl_down(d, off, 32);
    }
    if (lane == 0) { as_out[node] = s; ad_out[node] = d; }
}

// -------------------------------------------------------- edge kernels
__device__ __forceinline__ void atomicMaxF(float* addr, float val) {
    // IEEE-order trick: works for mixed-sign values, init must be very negative
    if (val >= 0.f) atomicMax((int*)addr, __float_as_int(val));
    else            atomicMin((unsigned int*)addr, __float_as_uint(val));
}

__global__ void edge_alpha_kernel(const int* __restrict__ src,
                                  const int* __restrict__ dst,
                                  const float* __restrict__ flepe,
                                  const float* __restrict__ wae,
                                  const float* __restrict__ asrc,
                                  const float* __restrict__ adst,
                                  float* __restrict__ alpha,
                                  float* __restrict__ amax) {
    int e = blockIdx.x * blockDim.x + threadIdx.x;
    if (e >= N_EDGES) return;
    const float* f = flepe + (size_t)e * EDGE_DIM;
    float ae = 0.f;
#pragma unroll
    for (int j = 0; j < EDGE_DIM; ++j) ae += f[j] * wae[j];
    int d = dst[e];
    float a = asrc[src[e]] + adst[d] + ae;
    a = (a > 0.f) ? a : NEG_SLOPE * a;      // leaky_relu
    alpha[e] = a;
    atomicMaxF(&amax[d], a);
}

__global__ void edge_exp_kernel(const int* __restrict__ dst,
                                const float* __restrict__ amax,
                                float* __restrict__ alpha_ex,   // alpha in, ex out
                                float* __restrict__ denom) {
    int e = blockIdx.x * blockDim.x + threadIdx.x;
    if (e >= N_EDGES) return;
    int d = dst[e];
    float ex = __expf(alpha_ex[e] - amax[d]);
    alpha_ex[e] = ex;
    atomicAdd(&denom[d], ex);
}

// out[dst,c] += (ex/denom[dst]) * h[src,c]  — one lane per (edge, channel)
template <int C>
__global__ void edge_aggregate_kernel(const int* __restrict__ src,
                                      const int* __restrict__ dst,
                                      const float* __restrict__ H,
                                      const float* __restrict__ exv,
                                      const float* __restrict__ denom,
                                      float* __restrict__ out) {
    const int epb = 256 / C;
    int e = blockIdx.x * epb + threadIdx.x / C;
    int c = threadIdx.x & (C - 1);
    if (e >= N_EDGES) return;
    int s = src[e], d = dst[e];
    float coef = exv[e] / (denom[d] + 1e-16f);
    atomicAdd(&out[(size_t)d * C + c], coef * H[(size_t)s * C + c]);
}

__global__ void bias_act_kernel(float* __restrict__ out, const float* __restrict__ b,
                                int C, int total, int do_relu) {
    int i = blockIdx.x * blockDim.x + threadIdx.x;
    if (i >= total) return;
    float v = out[i] + b[i & (C - 1)];
    out[i] = do_relu ? fmaxf(v, 0.f) : v;
}

// ---------------------------------------------------------------- launch
extern "C" void kernel_launch(void* const* d_in, const int* in_sizes, int n_in,
                              void* d_out, int out_size, void* d_ws, size_t ws_size,
                              hipStream_t stream) {
    const float* x     = (const float*)d_in[0];
    const int*   ei    = (const int*)  d_in[1];
    const float* flepe = (const float*)d_in[2];
    const float* W1    = (const float*)d_in[3];
    const float* as1   = (const float*)d_in[4];
    const float* ad1   = (const float*)d_in[5];
    const float* We1   = (const float*)d_in[6];
    const float* ae1   = (const float*)d_in[7];
    const float* b1    = (const float*)d_in[8];
    const float* W2    = (const float*)d_in[9];
    const float* as2   = (const float*)d_in[10];
    const float* ad2   = (const float*)d_in[11];
    const float* We2   = (const float*)d_in[12];
    const float* ae2   = (const float*)d_in[13];
    const float* b2    = (const float*)d_in[14];
    const int* src  = ei;
    const int* dstv = ei + N_EDGES;
    float* out = (float*)d_out;

    // workspace carve (~71 MB of fp32)
    float* ws    = (float*)d_ws;
    float* h1    = ws;                                  // N*128
    float* agg1  = h1   + (size_t)N_NODES * HID_CH;     // N*128 (layer-2 input)
    float* h2    = agg1 + (size_t)N_NODES * HID_CH;     // N*64
    float* asrc  = h2   + (size_t)N_NODES * OUT_CH;
    float* adst  = asrc + N_NODES;
    float* amax  = adst + N_NODES;
    float* denom = amax + N_NODES;
    float* alpha = denom + N_NODES;                     // E (alpha, then ex)
    float* wae1  = alpha + N_EDGES;                     // 16
    float* wae2  = wae1 + EDGE_DIM;                     // 16

    dim3 b256(256);
    dim3 gemm_block(32, 4);
    const dim3 gN((N_NODES + 255) / 256);
    const dim3 gE((N_EDGES + 255) / 256);
    const dim3 gNC128((N_NODES * HID_CH + 255) / 256);
    const dim3 gNC64((N_NODES * OUT_CH + 255) / 256);

    small_matvec_kernel<<<1, EDGE_DIM, 0, stream>>>(We1, ae1, wae1, HID_CH);
    small_matvec_kernel<<<1, EDGE_DIM, 0, stream>>>(We2, ae2, wae2, OUT_CH);

    // -------- layer 1: GATConv(x; W1) -> agg1, then ReLU --------
    fill_kernel<<<gN, b256, 0, stream>>>(amax, -3.0e38f, N_NODES);
    fill_kernel<<<gN, b256, 0, stream>>>(denom, 0.f, N_NODES);
    fill_kernel<<<gNC128, b256, 0, stream>>>(agg1, 0.f, N_NODES * HID_CH);

    gemm_wmma_f32<<<dim3(N_NODES / 16, HID_CH / 64), gemm_block, 0, stream>>>(
        x, W1, h1, IN_CH, HID_CH);
    node_alpha_kernel<<<dim3((N_NODES + 7) / 8), b256, 0, stream>>>(
        h1, as1, ad1, asrc, adst, HID_CH);
    edge_alpha_kernel<<<gE, b256, 0, stream>>>(src, dstv, flepe, wae1, asrc, adst,
                                               alpha, amax);
    edge_exp_kernel<<<gE, b256, 0, stream>>>(dstv, amax, alpha, denom);
    edge_aggregate_kernel<HID_CH><<<dim3(N_EDGES / 2), b256, 0, stream>>>(
        src, dstv, h1, alpha, denom, agg1);
    bias_act_kernel<<<gNC128, b256, 0, stream>>>(agg1, b1, HID_CH,
                                                 N_NODES * HID_CH, 1);

    // -------- layer 2: GATConv(relu(agg1); W2) -> d_out --------
    fill_kernel<<<gN, b256, 0, stream>>>(amax, -3.0e38f, N_NODES);
    fill_kernel<<<gN, b256, 0, stream>>>(denom, 0.f, N_NODES);
    fill_kernel<<<gNC64, b256, 0, stream>>>(out, 0.f, N_NODES * OUT_CH);

    gemm_wmma_f32<<<dim3(N_NODES / 16, OUT_CH / 64), gemm_block, 0, stream>>>(
        agg1, W2, h2, HID_CH, OUT_CH);
    node_alpha_kernel<<<dim3((N_NODES + 7) / 8), b256, 0, stream>>>(
        h2, as2, ad2, asrc, adst, OUT_CH);
    edge_alpha_kernel<<<gE, b256, 0, stream>>>(src, dstv, flepe, wae2, asrc, adst,
                                               alpha, amax);
    edge_exp_kernel<<<gE, b256, 0, stream>>>(dstv, amax, alpha, denom);
    edge_aggregate_kernel<OUT_CH><<<dim3(N_EDGES / 4), b256, 0, stream>>>(
        src, dstv, h2, alpha, denom, out);
    bias_act_kernel<<<gNC64, b256, 0, stream>>>(out, b2, OUT_CH,
                                                N_NODES * OUT_CH, 0);
}